// BiGraphGAT_25409026523339
// MI455X (gfx1250) — compile-verified
//
#include <hip/hip_runtime.h>
#include <cstdint>
#include <cstddef>

// ---------------------------------------------------------------------------
// BiGraphGAT for MI455X (gfx1250, wave32).
//   HD = 128 columns, H = 4 heads, D = 32 dims/head (fixed by reference).
//   GEMM via V_WMMA_F32_16X16X4_F32 (full f32 precision).
//   Edge phase: fp32 atomics (max_num_f32 / add_f32), coalesced 128-wide
//   gather/scatter per edge.
// ---------------------------------------------------------------------------

#define HD     128
#define NHEAD  4
#define NEGSLP 0.01f

typedef __attribute__((ext_vector_type(2))) float v2f;
typedef __attribute__((ext_vector_type(8))) float v8f;

// ---- GEMM: Y[n,j] = sum_k X[n,k] * W[j,k] + bias[j] -----------------------
// Block = 256 threads = 8 waves. Block handles 16 rows; wave w handles
// columns [16w, 16w+16). K-loop over 128 in steps of 4 (32 WMMAs per wave).
__global__ __launch_bounds__(256) void gat_linear_wmma(
    const float* __restrict__ X,    // [N, 128]
    const float* __restrict__ W,    // [128, 128] row-major (torch Linear weight)
    const float* __restrict__ bias, // [128]
    float* __restrict__ Y,          // [N, 128]
    int N)
{
  const int wave = threadIdx.x >> 5;   // 0..7 -> column tile
  const int lane = threadIdx.x & 31;
  const int g    = lane >> 4;          // half-wave select (K pair)
  const int l15  = lane & 15;

  int row0 = blockIdx.x * 16;
  if (row0 + 16 > N) row0 = N - 16;    // clamp (value-identical redundant work)
  const int col0 = wave * 16;
  const int ncol = col0 + l15;

  // A fragment source: row (row0 + l15), K = k + 2g, k+2g+1  (float2)
  const float* arow = X + (size_t)(row0 + l15) * HD;
  // B fragment: B[k][n] = W[n][k]; lane holds column ncol, K = k+2g, k+2g+1
  const float* bcol = W + (size_t)ncol * HD;

  v8f acc = {};
#pragma unroll
  for (int k = 0; k < HD; k += 4) {
    v2f a = *(const v2f*)(arow + k + 2 * g);
    v2f b = *(const v2f*)(bcol + k + 2 * g);
    acc = __builtin_amdgcn_wmma_f32_16x16x4_f32(
        /*neg_a=*/false, a, /*neg_b=*/false, b,
        /*c_mod=*/(short)0, acc, /*reuse_a=*/false, /*reuse_b=*/false);
  }

  // D layout: VGPR r -> M = r (lanes 0-15) / M = r+8 (lanes 16-31), N = l15.
  const float bn = bias[ncol];
#pragma unroll
  for (int r = 0; r < 8; ++r) {
    const int row = row0 + r + 8 * g;
    Y[(size_t)row * HD + ncol] = acc[r] + bn;
  }
}

// ---- per-node head dots: out[n,h] = sum_d Hf[n, h*32+d] * attn[h*32+d] ----
// One wave32 per node. Since D == 32, element (32*j + lane) belongs to head j.
__global__ __launch_bounds__(256) void gat_head_dot(
    const float* __restrict__ Hf,   // [N, 128]
    const float* __restrict__ attn, // [4*32]
    float* __restrict__ out,        // [N, 4]
    int N)
{
  const int node = (int)((blockIdx.x * (size_t)blockDim.x + threadIdx.x) >> 5);
  const int lane = threadIdx.x & 31;
  if (node >= N) return;
  const float* row = Hf + (size_t)node * HD;
  float p0 = row[lane      ] * attn[lane      ];
  float p1 = row[lane + 32 ] * attn[lane + 32 ];
  float p2 = row[lane + 64 ] * attn[lane + 64 ];
  float p3 = row[lane + 96 ] * attn[lane + 96 ];
#pragma unroll
  for (int off = 16; off > 0; off >>= 1) {
    p0 += __shfl_xor(p0, off, 32);
    p1 += __shfl_xor(p1, off, 32);
    p2 += __shfl_xor(p2, off, 32);
    p3 += __shfl_xor(p3, off, 32);
  }
  if (lane == 0) {
    out[node * 4 + 0] = p0;
    out[node * 4 + 1] = p1;
    out[node * 4 + 2] = p2;
    out[node * 4 + 3] = p3;
  }
}

// ---- init: m = -inf, s = 0, out = 0 --------------------------------------
__global__ void gat_init(float* __restrict__ mmax, float* __restrict__ ssum,
                         float* __restrict__ outp, int nmh, int nout)
{
  const int i = blockIdx.x * blockDim.x + threadIdx.x;
  if (i < nout) outp[i] = 0.0f;
  if (i < nmh) {
    mmax[i] = -__builtin_inff();
    ssum[i] = 0.0f;
  }
}

// ---- pass A: e = leaky_relu(el[src] + er[dst]); segment max over dst ------
__global__ __launch_bounds__(256) void gat_edge_score_max(
    const int* __restrict__ src, const int* __restrict__ dst,
    const float* __restrict__ el, const float* __restrict__ er,
    float* __restrict__ esc, float* __restrict__ mmax, int E)
{
  const int e = blockIdx.x * blockDim.x + threadIdx.x;
  if (e >= E) return;
  const int s = src[e], d = dst[e];
  const float4 l = ((const float4*)el)[s];
  const float4 r = ((const float4*)er)[d];
  float v[4] = {l.x + r.x, l.y + r.y, l.z + r.z, l.w + r.w};
#pragma unroll
  for (int h = 0; h < 4; ++h) {
    v[h] = v[h] > 0.0f ? v[h] : NEGSLP * v[h];
    __hip_atomic_fetch_max(&mmax[d * 4 + h], v[h],
                           __ATOMIC_RELAXED, __HIP_MEMORY_SCOPE_AGENT);
  }
  ((float4*)esc)[e] = make_float4(v[0], v[1], v[2], v[3]);
}

// ---- pass B: ex = exp(e - m[dst]); segment sum over dst -------------------
__global__ __launch_bounds__(256) void gat_edge_exp_sum(
    const int* __restrict__ dst, const float* __restrict__ mmax,
    float* __restrict__ esc, float* __restrict__ ssum, int E)
{
  const int e = blockIdx.x * blockDim.x + threadIdx.x;
  if (e >= E) return;
  const int d = dst[e];
  const float4 v = ((const float4*)esc)[e];
  const float4 m = ((const float4*)mmax)[d];
  const float ex0 = __expf(v.x - m.x);
  const float ex1 = __expf(v.y - m.y);
  const float ex2 = __expf(v.z - m.z);
  const float ex3 = __expf(v.w - m.w);
  ((float4*)esc)[e] = make_float4(ex0, ex1, ex2, ex3);
  atomicAdd(&ssum[d * 4 + 0], ex0);
  atomicAdd(&ssum[d * 4 + 1], ex1);
  atomicAdd(&ssum[d * 4 + 2], ex2);
  atomicAdd(&ssum[d * 4 + 3], ex3);
}

// ---- pass C: out[dst, c] += (ex / s[dst])[h(c)] * hs[src, c] --------------
// One block of 128 threads per edge: perfectly coalesced gather + scatter.
__global__ __launch_bounds__(128) void gat_aggregate(
    const int* __restrict__ src, const int* __restrict__ dst,
    const float* __restrict__ esc, const float* __restrict__ ssum,
    const float* __restrict__ hs, float* __restrict__ outp, int E)
{
  const int e = blockIdx.x;
  const int c = threadIdx.x;       // 0..127
  const int h = c >> 5;            // head = column / 32
  const int s = src[e], d = dst[e];
  const float a = esc[e * 4 + h] / ssum[d * 4 + h];
  atomicAdd(&outp[(size_t)d * HD + c], a * hs[(size_t)s * HD + c]);
}

// ---------------------------------------------------------------------------
extern "C" void kernel_launch(void* const* d_in, const int* in_sizes, int n_in,
                              void* d_out, int out_size, void* d_ws, size_t ws_size,
                              hipStream_t stream)
{
  const float* feat_src = (const float*)d_in[0];
  const float* feat_dst = (const float*)d_in[1];
  const int*   src_idx  = (const int*)d_in[2];
  const int*   dst_idx  = (const int*)d_in[3];
  const float* W_src    = (const float*)d_in[4];
  const float* b_src    = (const float*)d_in[5];
  const float* W_dst    = (const float*)d_in[6];
  const float* b_dst    = (const float*)d_in[7];
  const float* attn_l   = (const float*)d_in[8];
  const float* attn_r   = (const float*)d_in[9];

  const int N_SRC = in_sizes[0] / HD;
  const int N_DST = in_sizes[1] / HD;
  const int E     = in_sizes[2];

  // Workspace carve-out (256B aligned slabs).
  auto align256 = [](size_t x) { return (x + 255) & ~(size_t)255; };
  char* ws = (char*)d_ws;
  size_t off = 0;
  float* hs   = (float*)(ws + off); off += align256((size_t)N_SRC * HD * sizeof(float));
  float* hd   = (float*)(ws + off); off += align256((size_t)N_DST * HD * sizeof(float));
  float* el   = (float*)(ws + off); off += align256((size_t)N_SRC * NHEAD * sizeof(float));
  float* er   = (float*)(ws + off); off += align256((size_t)N_DST * NHEAD * sizeof(float));
  float* mmax = (float*)(ws + off); off += align256((size_t)N_DST * NHEAD * sizeof(float));
  float* ssum = (float*)(ws + off); off += align256((size_t)N_DST * NHEAD * sizeof(float));
  float* esc  = (float*)(ws + off); off += align256((size_t)E * NHEAD * sizeof(float));
  (void)off; (void)ws_size;

  float* outp = (float*)d_out;

  // 1) Linear projections (WMMA f32).
  gat_linear_wmma<<<(N_SRC + 15) / 16, 256, 0, stream>>>(feat_src, W_src, b_src, hs, N_SRC);
  gat_linear_wmma<<<(N_DST + 15) / 16, 256, 0, stream>>>(feat_dst, W_dst, b_dst, hd, N_DST);

  // 2) Per-node per-head attention dots.
  gat_head_dot<<<(N_SRC + 7) / 8, 256, 0, stream>>>(hs, attn_l, el, N_SRC);
  gat_head_dot<<<(N_DST + 7) / 8, 256, 0, stream>>>(hd, attn_r, er, N_DST);

  // 3) Init segment-max / segment-sum / output.
  const int nout = N_DST * HD;
  gat_init<<<(nout + 255) / 256, 256, 0, stream>>>(mmax, ssum, outp, N_DST * NHEAD, nout);

  // 4) Edge softmax + aggregation.
  gat_edge_score_max<<<(E + 255) / 256, 256, 0, stream>>>(src_idx, dst_idx, el, er, esc, mmax, E);
  gat_edge_exp_sum<<<(E + 255) / 256, 256, 0, stream>>>(dst_idx, mmax, esc, ssum, E);
  gat_aggregate<<<E, 128, 0, stream>>>(src_idx, dst_idx, esc, ssum, hs, outp, E);
}